// WarpAdjoint_31688268710628
// MI455X (gfx1250) — compile-verified
//
#include <hip/hip_runtime.h>

// Problem dims fixed by reference setup_inputs(): B=8, C=8, M=N=512.
#define BDIM 8
#define CDIM 8
#define MDIM 512
#define NDIM 512

#define TILE 64              // output tile owned per block
#define HALO 16              // 4-sigma halo for N(0, 4) flow
#define LTS  (TILE + 2*HALO) // 96  -> 96*96*4B = 36 KB LDS
#define NTHREADS 256

__global__ __launch_bounds__(NTHREADS)
void warp_adjoint_scatter(const float* __restrict__ x,
                          const float* __restrict__ u,
                          float* __restrict__ out)
{
    __shared__ float tile[LTS * LTS];

    const int tid = threadIdx.x;
    const int gj0 = blockIdx.x * TILE;
    const int gi0 = blockIdx.y * TILE;
    const int b   = blockIdx.z;

    // zero LDS accumulator
    for (int i = tid; i < LTS * LTS; i += NTHREADS) tile[i] = 0.0f;
    __syncthreads();

    // lane layout: 16 lanes per row, each lane handles 4 consecutive cols
    // -> one b128 x-load + two b128 u-loads per lane per pass (coalesced)
    const int colGroup  = (tid & 15) * 4;  // 0..60
    const int rowInPass = tid >> 4;        // 0..15

    const size_t plane = (size_t)MDIM * NDIM;
    float* const outb = out + (size_t)b * plane;

    for (int c = 0; c < CDIM; ++c) {
        const float* __restrict__ xbc = x + ((size_t)(b * CDIM + c)) * plane;
        const float* __restrict__ ubc = u + ((size_t)(b * CDIM + c)) * plane * 2;

        #pragma unroll
        for (int pass = 0; pass < 4; ++pass) {
            const int row = pass * 16 + rowInPass;  // 0..63
            const int gi  = gi0 + row;
            const int gj  = gj0 + colGroup;
            const size_t xoff = (size_t)gi * NDIM + gj;

            const float4 xv = *(const float4*)(xbc + xoff);
            const float4 ua = *(const float4*)(ubc + 2 * xoff);
            const float4 ub = *(const float4*)(ubc + 2 * xoff + 4);

            // prefetch next channel's same lines (global_prefetch_b8)
            if (c + 1 < CDIM) {
                __builtin_prefetch(xbc + plane + xoff, 0, 0);
                __builtin_prefetch(ubc + 2 * plane + 2 * xoff, 0, 0);
            }

            const float xs[4] = {xv.x, xv.y, xv.z, xv.w};
            const float ux[4] = {ua.x, ua.z, ub.x, ub.z};
            const float uy[4] = {ua.y, ua.w, ub.y, ub.w};

            #pragma unroll
            for (int k = 0; k < 4; ++k) {
                const float px = (float)(gj + k) + ux[k];
                const float py = (float)gi + uy[k];
                const float x0 = floorf(px);
                const float y0 = floorf(py);
                const float wx = px - x0;
                const float wy = py - y0;
                const int x0i = (int)x0;
                const int y0i = (int)y0;
                const float xval = xs[k];

                // local (tile) coords of the top-left neighbor
                const int lx0 = x0i - (gj0 - HALO);
                const int ly0 = y0i - (gi0 - HALO);

                const float w00 = (1.0f - wx) * (1.0f - wy);
                const float w10 = wx * (1.0f - wy);
                const float w01 = (1.0f - wx) * wy;
                const float w11 = wx * wy;

                if ((unsigned)lx0 < (unsigned)(LTS - 1) &&
                    (unsigned)ly0 < (unsigned)(LTS - 1)) {
                    // FAST PATH: whole 2x2 footprint inside the tile.
                    // No image-bounds checks needed: out-of-image cells are
                    // dropped at flush, matching the reference's zero-pad drop.
                    float* const p = &tile[ly0 * LTS + lx0];
                    atomicAdd(p,           w00 * xval);
                    atomicAdd(p + 1,       w10 * xval);
                    atomicAdd(p + LTS,     w01 * xval);
                    atomicAdd(p + LTS + 1, w11 * xval);
                } else {
                    // SLOW PATH (|u| beyond halo, ~4-sigma rare)
                    const float wv[4] = {w00, w10, w01, w11};
                    #pragma unroll
                    for (int n = 0; n < 4; ++n) {
                        const int dx = n & 1, dy = n >> 1;
                        const int xi = x0i + dx;
                        const int yi = y0i + dy;
                        const int lx = lx0 + dx;
                        const int ly = ly0 + dy;
                        const float v = wv[n] * xval;
                        if ((unsigned)lx < (unsigned)LTS &&
                            (unsigned)ly < (unsigned)LTS) {
                            atomicAdd(&tile[ly * LTS + lx], v);
                        } else if ((unsigned)xi < (unsigned)NDIM &&
                                   (unsigned)yi < (unsigned)MDIM) {
                            unsafeAtomicAdd(&outb[(size_t)yi * NDIM + xi], v);
                        }
                    }
                }
            }
        }
    }
    __syncthreads();

    // flush LDS accumulator with one global atomic per nonzero in-image cell
    for (int i = tid; i < LTS * LTS; i += NTHREADS) {
        const int ly = i / LTS;
        const int lx = i - ly * LTS;
        const int gy = gi0 - HALO + ly;
        const int gx = gj0 - HALO + lx;
        const float v = tile[i];
        if (v != 0.0f &&
            (unsigned)gy < (unsigned)MDIM && (unsigned)gx < (unsigned)NDIM) {
            unsafeAtomicAdd(&outb[(size_t)gy * NDIM + gx], v);
        }
    }
}

extern "C" void kernel_launch(void* const* d_in, const int* in_sizes, int n_in,
                              void* d_out, int out_size, void* d_ws, size_t ws_size,
                              hipStream_t stream) {
    (void)in_sizes; (void)n_in; (void)d_ws; (void)ws_size;
    const float* x = (const float*)d_in[0];
    const float* u = (const float*)d_in[1];
    float* out = (float*)d_out;

    // harness poisons d_out; we accumulate with atomics, so zero it first
    hipMemsetAsync(out, 0, (size_t)out_size * sizeof(float), stream);

    dim3 grid(NDIM / TILE, MDIM / TILE, BDIM);  // (8, 8, 8) = 512 blocks
    warp_adjoint_scatter<<<grid, NTHREADS, 0, stream>>>(x, u, out);
}